// MODEL_541165879955
// MI455X (gfx1250) — compile-verified
//
#include <hip/hip_runtime.h>
#include <math.h>

// Problem constants (from reference)
#define U_N  60000
#define I_N  120000
#define N_N  180000   // U + I
#define DD   32
#define DK   96
#define E_UU 960000
#define E_II 1920000
#define E_UI 1200000

typedef __attribute__((ext_vector_type(2))) float v2f;
typedef __attribute__((ext_vector_type(8))) float v8f;

// V_WMMA_F32_16X16X4_F32 : D(16x16,f32) = A(16x4,f32) x B(4x16,f32) + C
// 8-arg form: (neg_a, A, neg_b, B, c_mod, C, reuse_a, reuse_b)
__device__ __forceinline__ v8f wmma_k4(v2f a, v2f b, v8f c) {
  return __builtin_amdgcn_wmma_f32_16x16x4_f32(false, a, false, b, (short)0, c,
                                               false, false);
}

__device__ __forceinline__ float wave_sum(float v) {
#pragma unroll
  for (int o = 16; o; o >>= 1) v += __shfl_xor(v, o, 32);
  return v;
}
__device__ __forceinline__ float wave_max(float v) {
#pragma unroll
  for (int o = 16; o; o >>= 1) v = fmaxf(v, __shfl_xor(v, o, 32));
  return v;
}

// ---------------- degree / dinv ----------------
__global__ void deg_count_kernel(const int* __restrict__ idx, float* __restrict__ deg,
                                 int E, int off) {
  int t = blockIdx.x * blockDim.x + threadIdx.x;
  if (t < E) atomicAdd(&deg[idx[t] + off], 1.0f);
}

__global__ void dinv_kernel(float* __restrict__ deg, int n) {
  int t = blockIdx.x * blockDim.x + threadIdx.x;
  if (t < n) {
    float d = deg[t];
    deg[t] = (d > 0.0f) ? rsqrtf(d) : 0.0f;
  }
}

// ---------------- gate: out = emb * softmax(emb @ W + b, axis=1) ----------------
// wave per row; lane = output column
__global__ void gate_kernel(const float* __restrict__ emb, const float* __restrict__ W,
                            const float* __restrict__ b, float* __restrict__ out0,
                            float* __restrict__ sum0, int M) {
  int t = blockIdx.x * blockDim.x + threadIdx.x;
  int r = t >> 5, lane = t & 31;
  if (r >= M) return;
  size_t base = (size_t)r * DD;
  float x = emb[base + lane];
  float y = b[lane];
#pragma unroll
  for (int d = 0; d < DD; ++d) {
    float xd = __shfl(x, d, 32);
    y = fmaf(xd, W[d * DD + lane], y);
  }
  float m = wave_max(y);
  float e = __expf(y - m);
  float s = wave_sum(e);
  float g = x * (e / s);
  out0[base + lane] = g;
  sum0[base + lane] = g;
}

// ---------------- ui0 = concat(user_emb, item_emb); sum_ui = ui0 ----------------
__global__ void concat_init_kernel(const float* __restrict__ ue, const float* __restrict__ ie,
                                   float* __restrict__ ui0, float* __restrict__ sum_ui) {
  int t = blockIdx.x * blockDim.x + threadIdx.x;
  if (t >= N_N * DD) return;
  float v = (t < U_N * DD) ? ue[t] : ie[t - U_N * DD];
  ui0[t] = v;
  sum_ui[t] = v;
}

// ---------------- normalized SpMM: y[row] += dinv[row]*dinv[col]*x[col] ----------
// wave per edge, lane per feature dim
__global__ void spmm_kernel(const int* __restrict__ row, const int* __restrict__ col,
                            const float* __restrict__ dinv, const float* __restrict__ x,
                            float* __restrict__ y, int E, int roff, int coff) {
  int t = blockIdx.x * blockDim.x + threadIdx.x;
  int e = t >> 5, lane = t & 31;
  if (e >= E) return;
  int r = row[e] + roff;
  int c = col[e] + coff;
  float w = dinv[r] * dinv[c];
  float v = w * x[(size_t)c * DD + lane];
  atomicAdd(&y[(size_t)r * DD + lane], v);
}

// ---------------- per-layer combine ----------------
// uiEc[n] = (side0[n] + uie0[n]) / 2 ; sum_{u,i} += l2norm(side0) ; sum_ui += l2norm(uie0)
__global__ void combine_kernel(const float* __restrict__ ue0, const float* __restrict__ ie0,
                               const float* __restrict__ uie0, float* __restrict__ uiEc,
                               float* __restrict__ sum_u, float* __restrict__ sum_i,
                               float* __restrict__ sum_ui) {
  int t = blockIdx.x * blockDim.x + threadIdx.x;
  int n = t >> 5, lane = t & 31;
  if (n >= N_N) return;
  size_t base = (size_t)n * DD;
  float a = (n < U_N) ? ue0[base + lane] : ie0[(size_t)(n - U_N) * DD + lane];
  float c = uie0[base + lane];
  uiEc[base + lane] = 0.5f * (a + c);
  float na = wave_sum(a * a);
  float nc = wave_sum(c * c);
  float ia = 1.0f / fmaxf(sqrtf(na), 1e-12f);
  float ic = 1.0f / fmaxf(sqrtf(nc), 1e-12f);
  if (n < U_N) sum_u[base + lane] += a * ia;
  else         sum_i[(size_t)(n - U_N) * DD + lane] += a * ia;
  sum_ui[base + lane] += c * ic;
}

__global__ void scale_kernel(float* __restrict__ p, float s, int n) {
  int t = blockIdx.x * blockDim.x + threadIdx.x;
  if (t < n) p[t] *= s;
}

// ---------------- neighbor sums over UI edges ----------------
__global__ void nb_kernel(const int* __restrict__ ui_u, const int* __restrict__ ui_i,
                          const float* __restrict__ ui_user, const float* __restrict__ ui_item,
                          float* __restrict__ unb, float* __restrict__ inb, int E) {
  int t = blockIdx.x * blockDim.x + threadIdx.x;
  int e = t >> 5, lane = t & 31;
  if (e >= E) return;
  int u = ui_u[e];
  int it = ui_i[e];
  atomicAdd(&unb[(size_t)u * DD + lane], ui_item[(size_t)it * DD + lane]);
  atomicAdd(&inb[(size_t)it * DD + lane], ui_user[(size_t)u * DD + lane]);
}

// ---------------- tembed = concat(emb, uix, nb, nb) @ Wm + bias (WMMA fp32) ------
// block = 4 waves, wave per 16-row tile; B (128x32) staged in LDS
#define TB_WAVES 4
__global__ __launch_bounds__(128) void tembed_kernel(
    const float* __restrict__ emb, const float* __restrict__ uix,
    const float* __restrict__ nb, const float* __restrict__ Wm,
    const float* __restrict__ bias, float* __restrict__ tout, int M) {
  __shared__ float sW[128 * DD];
  int tid = threadIdx.x;
  for (int t = tid; t < 128 * DD; t += 128) sW[t] = Wm[t];
  __syncthreads();
  int wave = tid >> 5, lane = tid & 31;
  int tile = blockIdx.x * TB_WAVES + wave;
  int row0 = tile * 16;
  if (row0 >= M) return;
  int half = lane >> 4;      // A: lanes 0-15 hold K=0,1; lanes 16-31 hold K=2,3
  int mrow = lane & 15;
  v8f c0 = {};
  v8f c1 = {};
  const float* srcs[4] = {emb, uix, nb, nb};
#pragma unroll
  for (int s = 0; s < 4; ++s) {
    const float* src = srcs[s] + (size_t)(row0 + mrow) * DD;
#pragma unroll
    for (int kk = 0; kk < 8; ++kk) {
      int kb = kk * 4 + half * 2;
      v2f a;
      a.x = src[kb];
      a.y = src[kb + 1];
      int kg = s * DD + kb;   // global K index of this lane's first B row
      v2f b0, b1;
      b0.x = sW[kg * DD + mrow];
      b0.y = sW[(kg + 1) * DD + mrow];
      b1.x = sW[kg * DD + 16 + mrow];
      b1.y = sW[(kg + 1) * DD + 16 + mrow];
      c0 = wmma_k4(a, b0, c0);
      c1 = wmma_k4(a, b1, c1);
    }
  }
#pragma unroll
  for (int v = 0; v < 8; ++v) {
    int r = row0 + half * 8 + v;  // D: vgpr v -> M = v (+8 for lanes 16-31)
    tout[(size_t)r * DD + mrow] = c0[v] + bias[mrow];
    tout[(size_t)r * DD + 16 + mrow] = c1[v] + bias[16 + mrow];
  }
}

// ---------------- meta MLP: o = l2norm(prelu(x@Wp+bp) @ Wo + bo) (WMMA fp32) -----
// block = 2 waves; Wp (12KB) + Wo (36KB) + h tiles (12KB) + partial(384B) in LDS
#define ML_WAVES 2
__global__ __launch_bounds__(64) void mlp_kernel(
    const float* __restrict__ x, const float* __restrict__ Wp, const float* __restrict__ bp,
    const float* __restrict__ Wo, const float* __restrict__ bo,
    float* __restrict__ meta, float* __restrict__ meansum, int M) {
  __shared__ float sWp[DD * DK];
  __shared__ float sWo[DK * DK];
  __shared__ float sH[ML_WAVES][16 * DK];
  __shared__ float sPart[DK];
  int tid = threadIdx.x;
  for (int t = tid; t < DD * DK; t += 64) sWp[t] = Wp[t];
  for (int t = tid; t < DK * DK; t += 64) sWo[t] = Wo[t];
  for (int t = tid; t < DK; t += 64) sPart[t] = 0.0f;
  __syncthreads();
  int wave = tid >> 5, lane = tid & 31;
  int half = lane >> 4, mrow = lane & 15;
  int row0 = (blockIdx.x * ML_WAVES + wave) * 16;  // M % 32 == 0: always full
  float* hme = sH[wave];

  // GEMM1: x(16x32) @ Wp(32x96)
  v8f acc[6];
#pragma unroll
  for (int i = 0; i < 6; ++i) acc[i] = (v8f){};
  const float* xr = x + (size_t)(row0 + mrow) * DD;
#pragma unroll
  for (int kk = 0; kk < 8; ++kk) {
    int kb = kk * 4 + half * 2;
    v2f a;
    a.x = xr[kb];
    a.y = xr[kb + 1];
#pragma unroll
    for (int nt = 0; nt < 6; ++nt) {
      int n = nt * 16 + mrow;
      v2f b;
      b.x = sWp[kb * DK + n];
      b.y = sWp[(kb + 1) * DK + n];
      acc[nt] = wmma_k4(a, b, acc[nt]);
    }
  }
  // bias + PReLU -> LDS h tile (same-wave LDS ops are in-order)
#pragma unroll
  for (int nt = 0; nt < 6; ++nt) {
    int n = nt * 16 + mrow;
    float bpn = bp[n];
#pragma unroll
    for (int v = 0; v < 8; ++v) {
      float val = acc[nt][v] + bpn;
      val = (val >= 0.0f) ? val : 0.25f * val;
      hme[(half * 8 + v) * DK + n] = val;
    }
  }
  // GEMM2: h(16x96) @ Wo(96x96)
  v8f acc2[6];
#pragma unroll
  for (int i = 0; i < 6; ++i) acc2[i] = (v8f){};
#pragma unroll
  for (int kk = 0; kk < 24; ++kk) {
    int kb = kk * 4 + half * 2;
    v2f a;
    a.x = hme[mrow * DK + kb];
    a.y = hme[mrow * DK + kb + 1];
#pragma unroll
    for (int nt = 0; nt < 6; ++nt) {
      int n = nt * 16 + mrow;
      v2f b;
      b.x = sWo[kb * DK + n];
      b.y = sWo[(kb + 1) * DK + n];
      acc2[nt] = wmma_k4(a, b, acc2[nt]);
    }
  }
  // bias, stash o tile into LDS (h fully consumed)
#pragma unroll
  for (int nt = 0; nt < 6; ++nt) {
    int n = nt * 16 + mrow;
    float bon = bo[n];
#pragma unroll
    for (int v = 0; v < 8; ++v) hme[(half * 8 + v) * DK + n] = acc2[nt][v] + bon;
  }
  // row-wise l2norm, write meta, accumulate block-local column sums
  for (int r = 0; r < 16; ++r) {
    float o0 = hme[r * DK + lane];
    float o1 = hme[r * DK + 32 + lane];
    float o2 = hme[r * DK + 64 + lane];
    float ss = wave_sum(o0 * o0 + o1 * o1 + o2 * o2);
    float inv = 1.0f / fmaxf(sqrtf(ss), 1e-12f);
    float n0 = o0 * inv, n1 = o1 * inv, n2 = o2 * inv;
    size_t base = (size_t)(row0 + r) * DK;
    meta[base + lane] = n0;
    meta[base + 32 + lane] = n1;
    meta[base + 64 + lane] = n2;
    atomicAdd(&sPart[lane], n0);
    atomicAdd(&sPart[32 + lane], n1);
    atomicAdd(&sPart[64 + lane], n2);
  }
  __syncthreads();
  for (int t = tid; t < DK; t += 64) atomicAdd(&meansum[t], sPart[t]);
}

__global__ void mean_div_kernel(float* __restrict__ means) {
  int t = blockIdx.x * blockDim.x + threadIdx.x;
  if (t < 2 * DK) means[t] *= (1.0f / (float)U_N);
  else if (t < 4 * DK) means[t] *= (1.0f / (float)I_N);
}

// ---------------- final per-row meta transform ----------------
// wave per row; lane = feature dim d
__global__ void final_kernel(const float* __restrict__ emb, const float* __restrict__ uix,
                             const float* __restrict__ meta1, const float* __restrict__ meta2,
                             const float* __restrict__ mean1, const float* __restrict__ mean2,
                             float* __restrict__ outF, float* __restrict__ outP,
                             float* __restrict__ outUi, int M) {
  int t = blockIdx.x * blockDim.x + threadIdx.x;
  int r = t >> 5, lane = t & 31;
  if (r >= M) return;
  size_t b32 = (size_t)r * DD;
  float e = emb[b32 + lane];
  float xw = uix[b32 + lane];
  float tu = e * xw;
  const float* m1 = meta1 + (size_t)r * DK;
  const float* m2 = meta2 + (size_t)r * DK;
  // lw1: (D,K) layout row[d*3+k]; softmax over d; t1[k] = sum_d tu_d * lw1[d,k]
  float t1[3];
#pragma unroll
  for (int k = 0; k < 3; ++k) {
    float z = m1[lane * 3 + k] + mean1[lane * 3 + k];
    float mx = wave_max(z);
    float ez = __expf(z - mx);
    float s = wave_sum(ez);
    t1[k] = wave_sum(tu * (ez / s));
  }
  // lw2: (K,D) layout row[k*32+d]; softmax over k (per-lane local)
  float z0 = m2[lane] + mean2[lane];
  float z1 = m2[32 + lane] + mean2[32 + lane];
  float z2 = m2[64 + lane] + mean2[64 + lane];
  float mx = fmaxf(z0, fmaxf(z1, z2));
  float e0 = __expf(z0 - mx), e1 = __expf(z1 - mx), e2 = __expf(z2 - mx);
  float s = e0 + e1 + e2;
  float t2 = (t1[0] * e0 + t1[1] * e1 + t1[2] * e2) / s;
  float F = e + t2;
  outF[b32 + lane] = F;
  outP[b32 + lane] = 0.8f * xw + 0.2f * F;
  outUi[b32 + lane] = xw;
}

static inline int cdiv(int a, int b) { return (a + b - 1) / b; }

extern "C" void kernel_launch(void* const* d_in, const int* in_sizes, int n_in,
                              void* d_out, int out_size, void* d_ws, size_t ws_size,
                              hipStream_t stream) {
  (void)in_sizes; (void)n_in; (void)out_size; (void)ws_size;
  const float* user_emb = (const float*)d_in[0];
  const float* item_emb = (const float*)d_in[1];
  const float* gwu  = (const float*)d_in[2];
  const float* gwub = (const float*)d_in[3];
  const float* gwi  = (const float*)d_in[4];
  const float* gwib = (const float*)d_in[5];
  const float* Wmu  = (const float*)d_in[6];
  const float* bmu  = (const float*)d_in[7];
  const float* Wmi  = (const float*)d_in[8];
  const float* bmi  = (const float*)d_in[9];
  // mlps flattened: 4 x (Wp, bp, Wo, bo)
  const float* mW[4][4];
  for (int m = 0; m < 4; ++m)
    for (int j = 0; j < 4; ++j) mW[m][j] = (const float*)d_in[10 + m * 4 + j];
  const int* uu_row = (const int*)d_in[26];
  const int* uu_col = (const int*)d_in[27];
  const int* ii_row = (const int*)d_in[28];
  const int* ii_col = (const int*)d_in[29];
  const int* ui_u   = (const int*)d_in[30];
  const int* ui_i   = (const int*)d_in[31];
  float* out = (float*)d_out;

  // ---- workspace layout (floats) ----
  float* ws = (float*)d_ws;
  float* deg    = ws;                         // U+I+N (deg -> dinv in place)
  float* dinv_uu = deg;
  float* dinv_ii = deg + U_N;
  float* dinv_ui = deg + U_N + I_N;
  float* uu0    = deg + (U_N + I_N + N_N);    // U*32 (later reused as unb)
  float* ii0    = uu0 + (size_t)U_N * DD;     // I*32 (later reused as inb)
  float* ui0    = ii0 + (size_t)I_N * DD;     // N*32 (later reused as tembedu|tembedi)
  float* uiEc   = ui0 + (size_t)N_N * DD;     // N*32
  float* ue0    = uiEc + (size_t)N_N * DD;    // U*32 (ue0,ie0,uie0 contiguous: one memset)
  float* ie0    = ue0 + (size_t)U_N * DD;     // I*32
  float* uie0   = ie0 + (size_t)I_N * DD;     // N*32
  float* sum_u  = uie0 + (size_t)N_N * DD;    // U*32 (sum_u,sum_i,sum_ui contiguous: one scale)
  float* sum_i  = sum_u + (size_t)U_N * DD;   // I*32
  float* sum_ui = sum_i + (size_t)I_N * DD;   // N*32
  float* metau1 = sum_ui + (size_t)N_N * DD;  // U*96
  float* metau2 = metau1 + (size_t)U_N * DK;  // U*96
  float* metai1 = metau2 + (size_t)U_N * DK;  // I*96
  float* metai2 = metai1 + (size_t)I_N * DK;  // I*96
  float* means  = metai2 + (size_t)I_N * DK;  // 4*96

  const int TPB = 256;

  // ---- degrees -> dinv ----
  (void)hipMemsetAsync(deg, 0, sizeof(float) * (size_t)(U_N + I_N + N_N), stream);
  (void)hipMemsetAsync(means, 0, sizeof(float) * 4 * DK, stream);
  deg_count_kernel<<<cdiv(E_UU, TPB), TPB, 0, stream>>>(uu_row, dinv_uu, E_UU, 0);
  deg_count_kernel<<<cdiv(E_II, TPB), TPB, 0, stream>>>(ii_row, dinv_ii, E_II, 0);
  deg_count_kernel<<<cdiv(E_UI, TPB), TPB, 0, stream>>>(ui_u, dinv_ui, E_UI, 0);
  deg_count_kernel<<<cdiv(E_UI, TPB), TPB, 0, stream>>>(ui_i, dinv_ui, E_UI, U_N);
  dinv_kernel<<<cdiv(U_N + I_N + N_N, TPB), TPB, 0, stream>>>(deg, U_N + I_N + N_N);

  // ---- layer-0 features ----
  gate_kernel<<<cdiv(U_N * 32, TPB), TPB, 0, stream>>>(user_emb, gwu, gwub, uu0, sum_u, U_N);
  gate_kernel<<<cdiv(I_N * 32, TPB), TPB, 0, stream>>>(item_emb, gwi, gwib, ii0, sum_i, I_N);
  concat_init_kernel<<<cdiv(N_N * DD, TPB), TPB, 0, stream>>>(user_emb, item_emb, ui0, sum_ui);

  // ---- propagation layers ----
  for (int layer = 0; layer < 2; ++layer) {
    const float* xu  = (layer == 0) ? uu0 : uiEc;
    const float* xi  = (layer == 0) ? ii0 : (uiEc + (size_t)U_N * DD);
    const float* xui = (layer == 0) ? ui0 : uiEc;
    (void)hipMemsetAsync(ue0, 0, sizeof(float) * (size_t)(U_N + I_N + N_N) * DD, stream);
    spmm_kernel<<<cdiv(E_UU * 32, TPB), TPB, 0, stream>>>(uu_row, uu_col, dinv_uu, xu, ue0, E_UU, 0, 0);
    spmm_kernel<<<cdiv(E_II * 32, TPB), TPB, 0, stream>>>(ii_row, ii_col, dinv_ii, xi, ie0, E_II, 0, 0);
    spmm_kernel<<<cdiv(E_UI * 32, TPB), TPB, 0, stream>>>(ui_u, ui_i, dinv_ui, xui, uie0, E_UI, 0, U_N);
    spmm_kernel<<<cdiv(E_UI * 32, TPB), TPB, 0, stream>>>(ui_i, ui_u, dinv_ui, xui, uie0, E_UI, U_N, 0);
    combine_kernel<<<cdiv(N_N * 32, TPB), TPB, 0, stream>>>(ue0, ie0, uie0, uiEc, sum_u, sum_i, sum_ui);
  }
  // means over 3 stacked layers
  scale_kernel<<<cdiv((U_N + I_N + N_N) * DD, TPB), TPB, 0, stream>>>(
      sum_u, 1.0f / 3.0f, (U_N + I_N + N_N) * DD);
  // sum_u = userEmbedding, sum_i = itemEmbedding, sum_ui = uiEmbedding
  const float* ui_user = sum_ui;
  const float* ui_item = sum_ui + (size_t)U_N * DD;

  // ---- neighbor aggregates (reuse uu0 -> unb, ii0 -> inb) ----
  float* unb = uu0;
  float* inb = ii0;
  (void)hipMemsetAsync(unb, 0, sizeof(float) * (size_t)(U_N + I_N) * DD, stream);
  nb_kernel<<<cdiv(E_UI * 32, TPB), TPB, 0, stream>>>(ui_u, ui_i, ui_user, ui_item, unb, inb, E_UI);

  // ---- tembed GEMMs (WMMA fp32, K=128 N=32) ; reuse ui0 buffer ----
  float* tembedu = ui0;
  float* tembedi = ui0 + (size_t)U_N * DD;
  tembed_kernel<<<cdiv(U_N / 16, TB_WAVES), 128, 0, stream>>>(sum_u, ui_user, unb, Wmu, bmu, tembedu, U_N);
  tembed_kernel<<<cdiv(I_N / 16, TB_WAVES), 128, 0, stream>>>(sum_i, ui_item, inb, Wmi, bmi, tembedi, I_N);

  // ---- meta MLPs (WMMA fp32) ----
  mlp_kernel<<<U_N / 32, 64, 0, stream>>>(tembedu, mW[0][0], mW[0][1], mW[0][2], mW[0][3],
                                          metau1, means + 0 * DK, U_N);
  mlp_kernel<<<U_N / 32, 64, 0, stream>>>(tembedu, mW[1][0], mW[1][1], mW[1][2], mW[1][3],
                                          metau2, means + 1 * DK, U_N);
  mlp_kernel<<<I_N / 32, 64, 0, stream>>>(tembedi, mW[2][0], mW[2][1], mW[2][2], mW[2][3],
                                          metai1, means + 2 * DK, I_N);
  mlp_kernel<<<I_N / 32, 64, 0, stream>>>(tembedi, mW[3][0], mW[3][1], mW[3][2], mW[3][3],
                                          metai2, means + 3 * DK, I_N);
  mean_div_kernel<<<cdiv(4 * DK, TPB), TPB, 0, stream>>>(means);

  // ---- final transform + outputs ----
  // out rows: [uF(U), iF(I), pu(U), pi(I), ui_user(U), ui_item(I)] x 32
  float* out_uF = out;
  float* out_iF = out + (size_t)U_N * DD;
  float* out_pu = out + (size_t)(U_N + I_N) * DD;
  float* out_pi = out + (size_t)(2 * U_N + I_N) * DD;
  float* out_uu = out + (size_t)(2 * U_N + 2 * I_N) * DD;
  float* out_ii = out + (size_t)(3 * U_N + 2 * I_N) * DD;
  final_kernel<<<cdiv(U_N * 32, TPB), TPB, 0, stream>>>(sum_u, ui_user, metau1, metau2,
                                                        means + 0 * DK, means + 1 * DK,
                                                        out_uF, out_pu, out_uu, U_N);
  final_kernel<<<cdiv(I_N * 32, TPB), TPB, 0, stream>>>(sum_i, ui_item, metai1, metai2,
                                                        means + 2 * DK, means + 3 * DK,
                                                        out_iF, out_pi, out_ii, I_N);
}